// SmoothLoss_70437463654700
// MI455X (gfx1250) — compile-verified
//
#include <hip/hip_runtime.h>
#include <hip/hip_bf16.h>

// Problem constants (from reference)
#define NB   4
#define HH_  512
#define WW_  512
#define IH_  1024
#define IW_  1024
#define BLOCKS2 1024
#define THREADS2 256
#define TOTAL_PIX (NB * HH_ * WW_)                 // 1,048,576
#define INV_COUNT (1.0f / 8388608.0f)              // 1 / (N*8*H*W)

typedef __attribute__((ext_vector_type(2))) float v2f;
typedef __attribute__((ext_vector_type(8))) float v8f;

// Exact f32 wave32 reduction via V_WMMA_F32_16X16X4_F32:
// A holds the 32 per-lane partials (a.y = 0), B = all ones, so
// D[m][n] = sum_k A[m][k]. Column 0 lives in lane 0 (m=0..7) and
// lane 16 (m=8..15); summing those 8-vectors yields the wave total.
// EXEC must be all ones at the call site.
__device__ __forceinline__ float wave_sum_wmma(float x) {
    v2f a; a.x = x;    a.y = 0.0f;
    v2f b; b.x = 1.0f; b.y = 1.0f;
    v8f c = {};
    c = __builtin_amdgcn_wmma_f32_16x16x4_f32(false, a, false, b,
                                              (short)0, c, false, false);
    float s = c[0] + c[1] + c[2] + c[3] + c[4] + c[5] + c[6] + c[7];
    return __shfl(s, 0, 32) + __shfl(s, 16, 32);
}

// Pass 1: fused 2x bilinear downsample (== 2x2 box average for exact 2x,
// half-pixel centers) + RGB->YUV. image [N,3,1024,1024] -> yuv [N,3,512,512].
__global__ void yuv_downsample_kernel(const float* __restrict__ img,
                                      float* __restrict__ yuv) {
    int idx = blockIdx.x * blockDim.x + threadIdx.x;
    if (idx >= TOTAL_PIX) return;
    int w = idx % WW_;
    int h = (idx / WW_) % HH_;
    int n = idx / (WW_ * HH_);

    const float* base = img + (size_t)n * 3 * IH_ * IW_;
    float ch[3];
#pragma unroll
    for (int cci = 0; cci < 3; ++cci) {
        const float* p = base + (size_t)cci * IH_ * IW_
                              + (size_t)(2 * h) * IW_ + (size_t)(2 * w);
        ch[cci] = 0.25f * (p[0] + p[1] + p[IW_] + p[IW_ + 1]);
    }
    float y =  0.299f * ch[0] + 0.587f * ch[1] + 0.114f * ch[2];
    float u = -0.147f * ch[0] - 0.289f * ch[1] + 0.436f * ch[2];
    float v =  0.615f * ch[0] - 0.515f * ch[1] - 0.100f * ch[2];

    size_t o = (size_t)n * 3 * HH_ * WW_ + (size_t)h * WW_ + (size_t)w;
    yuv[o]                        = y;
    yuv[o + (size_t)HH_ * WW_]     = u;
    yuv[o + (size_t)2 * HH_ * WW_] = v;
}

// Pass 2: bilateral affinity * pred-diff^2, summed per block (deterministic:
// fixed grid, fixed intra-block order). Neighbor offsets are {-2,0,2}^2 minus
// center with replicate clamping (matches edge-padded unfold of both values
// and positions).
__global__ void loss_partial_kernel(const float* __restrict__ pred,
                                    const float* __restrict__ yuv,
                                    float* __restrict__ partials) {
    const int tid    = blockIdx.x * blockDim.x + threadIdx.x;
    const int stride = gridDim.x * blockDim.x;   // 262,144: divides TOTAL_PIX
    const int HW = HH_ * WW_;

    float acc = 0.0f;
    for (int idx = tid; idx < TOTAL_PIX; idx += stride) {
        int w = idx % WW_;
        int h = (idx / WW_) % HH_;
        int n = idx / HW;

        const float* yb = yuv  + (size_t)n * 3 * HW;
        const float* pb = pred + (size_t)n * HW;
        int o = h * WW_ + w;
        float y0 = yb[o];
        float u0 = yb[o + HW];
        float v0 = yb[o + 2 * HW];
        float p0 = pb[o];

#pragma unroll
        for (int di = -1; di <= 1; ++di) {
#pragma unroll
            for (int dj = -1; dj <= 1; ++dj) {
                if (di == 0 && dj == 0) continue;
                int hn = min(max(h + 2 * di, 0), HH_ - 1);
                int wn = min(max(w + 2 * dj, 0), WW_ - 1);
                float dr = (float)(h - hn);
                float dc = (float)(w - wn);
                int oo = hn * WW_ + wn;
                float dy = y0 - yb[oo];
                float du = u0 - yb[oo + HW];
                float dv = v0 - yb[oo + 2 * HW];
                // pos/(16*2) + luma/(16*2) + chroma/(8*2)
                float t = (dr * dr + dc * dc + dy * dy) * (1.0f / 32.0f)
                        + (du * du + dv * dv) * (1.0f / 16.0f);
                float aff = expf(-t);
                float pd  = p0 - pb[oo];
                acc += aff * pd * pd;
            }
        }
    }

    // Per-wave WMMA reduction (all lanes active here), then fixed-order
    // cross-wave sum in LDS by thread 0.
    float wsum = wave_sum_wmma(acc);
    __shared__ float smem[THREADS2 / 32];
    int lane = threadIdx.x & 31;
    int wid  = threadIdx.x >> 5;
    if (lane == 0) smem[wid] = wsum;
    __syncthreads();
    if (threadIdx.x == 0) {
        float b = 0.0f;
#pragma unroll
        for (int i = 0; i < THREADS2 / 32; ++i) b += smem[i];
        partials[blockIdx.x] = b;
    }
}

// Final: one wave sums the 1024 block partials (fixed order per lane),
// WMMA-reduces across the wave, scales by 1/(N*K*H*W).
__global__ void finalize_kernel(const float* __restrict__ partials,
                                float* __restrict__ out) {
    float acc = 0.0f;
    for (int i = threadIdx.x; i < BLOCKS2; i += 32) acc += partials[i];
    float total = wave_sum_wmma(acc);
    if (threadIdx.x == 0) out[0] = total * INV_COUNT;
}

extern "C" void kernel_launch(void* const* d_in, const int* in_sizes, int n_in,
                              void* d_out, int out_size, void* d_ws, size_t ws_size,
                              hipStream_t stream) {
    const float* pred  = (const float*)d_in[0];
    // d_in[1] (target) is unused by the reference loss.
    const float* image = (const float*)d_in[2];
    float* out = (float*)d_out;

    // Workspace layout: [ YUV buffer: N*3*H*W floats (12 MB) | partials: 1024 floats ]
    float* yuv      = (float*)d_ws;
    float* partials = yuv + (size_t)NB * 3 * HH_ * WW_;

    yuv_downsample_kernel<<<(TOTAL_PIX + 255) / 256, 256, 0, stream>>>(image, yuv);
    loss_partial_kernel<<<BLOCKS2, THREADS2, 0, stream>>>(pred, yuv, partials);
    finalize_kernel<<<1, 32, 0, stream>>>(partials, out);
}